// GATLayer_29918742184382
// MI455X (gfx1250) — compile-verified
//
#include <hip/hip_runtime.h>

#define NNODES 50000
#define NEDGES 800000
#define DIM 64
#define NEG_SLOPE 0.01f

typedef float v2f __attribute__((ext_vector_type(2)));
typedef float v8f __attribute__((ext_vector_type(8)));

// Order-preserving map float -> uint so we can use atomicMax(uint) for float max.
__device__ __forceinline__ unsigned f2ord(float f) {
    unsigned u = __float_as_uint(f);
    return (u & 0x80000000u) ? ~u : (u | 0x80000000u);
}
__device__ __forceinline__ float ord2f(unsigned u) {
    return __uint_as_float((u & 0x80000000u) ? (u & 0x7FFFFFFFu) : ~u);
}
#define ORD_NEG_INF 0x007FFFFFu   // f2ord(-inf)

// ---------------- Kernel A: per-node projections + segment-state init ----------------
__global__ void __launch_bounds__(256)
gat_node_proj(const float* __restrict__ z_src, const float* __restrict__ z_t,
              const float* __restrict__ w,
              float* __restrict__ p1, float* __restrict__ p2,
              unsigned* __restrict__ segMaxU, float* __restrict__ segSum) {
    int n = blockIdx.x * blockDim.x + threadIdx.x;
    if (n >= NNODES) return;
    const float4* a  = (const float4*)(z_src + (size_t)n * DIM);
    const float4* b  = (const float4*)(z_t   + (size_t)n * DIM);
    const float4* w1 = (const float4*)(w);
    const float4* w2 = (const float4*)(w + DIM);
    float s1 = 0.f, s2 = 0.f;
#pragma unroll
    for (int i = 0; i < DIM / 4; ++i) {
        float4 av = a[i], wv = w1[i];
        s1 = fmaf(av.x, wv.x, s1); s1 = fmaf(av.y, wv.y, s1);
        s1 = fmaf(av.z, wv.z, s1); s1 = fmaf(av.w, wv.w, s1);
        float4 bv = b[i], uv = w2[i];
        s2 = fmaf(bv.x, uv.x, s2); s2 = fmaf(bv.y, uv.y, s2);
        s2 = fmaf(bv.z, uv.z, s2); s2 = fmaf(bv.w, uv.w, s2);
    }
    p1[n] = s1;
    p2[n] = s2;
    segMaxU[n] = ORD_NEG_INF;
    segSum[n]  = 0.f;
}

// ---------------- Kernel B: WMMA edge scores (16 edges per wave) ----------------
// A-tile (16x4 f32) per V_WMMA_F32_16X16X4_F32 layout:
//   lanes 0-15  : row M=lane,    values K = k0+0, k0+1  (VGPR0, VGPR1)
//   lanes 16-31 : row M=lane-16, values K = k0+2, k0+3
// B-tile (4x16): every column = w3[k0..k0+3] -> all 16 output columns identical.
// C/D f32 16x16 layout: lane 0 holds D[0..7][0] in acc[0..7], lane 16 holds D[8..15][0].
__global__ void __launch_bounds__(256)
gat_edge_score(const float* __restrict__ zc, const float* __restrict__ w3,
               const float* __restrict__ p1, const float* __restrict__ p2,
               const int* __restrict__ src, const int* __restrict__ dst,
               float* __restrict__ aout, unsigned* __restrict__ segMaxU) {
    const int tid  = threadIdx.x;
    const int lane = tid & 31;
    const int half = lane >> 4;     // 0: K pair {0,1}, 1: K pair {2,3}
    const int lid  = lane & 15;     // row within the 16-edge tile
    const int wave = tid >> 5;
    const int e0   = (blockIdx.x * 8 + wave) * 16;   // NEDGES % 128 == 0 -> no tail

    const float* rowPtr = zc + (size_t)(e0 + lid) * DIM + 2 * half;

    v8f acc = {0.f, 0.f, 0.f, 0.f, 0.f, 0.f, 0.f, 0.f};
#pragma unroll
    for (int s = 0; s < DIM / 4; ++s) {
        v2f av = *(const v2f*)(rowPtr + 4 * s);           // global_load_b64, streams z_c
        v2f bv = *(const v2f*)(w3 + 4 * s + 2 * half);    // L0-resident weight pair
        acc = __builtin_amdgcn_wmma_f32_16x16x4_f32(
            /*neg_a=*/false, av, /*neg_b=*/false, bv,
            /*c_mod=*/(short)0, acc, /*reuse_a=*/false, /*reuse_b=*/false);
    }

    if (lid == 0) {  // lanes 0 and 16 own column 0 of the result tile
        const int mbase = half * 8;
#pragma unroll
        for (int r = 0; r < 8; ++r) {
            const int e = e0 + mbase + r;
            const int si = src[e];
            const int di = dst[e];
            float aval = acc[r] + p1[si] + p2[di];
            aval = (aval > 0.f) ? aval : aval * NEG_SLOPE;
            aout[e] = aval;
            atomicMax(&segMaxU[di], f2ord(aval));
        }
    }
}

// ---------------- Kernel C: e = exp(a - max[dst]); segSum += e ----------------
__global__ void __launch_bounds__(256)
gat_edge_exp(const int* __restrict__ dst, const unsigned* __restrict__ segMaxU,
             float* __restrict__ out, float* __restrict__ segSum) {
    int i = blockIdx.x * blockDim.x + threadIdx.x;
    if (i >= NEDGES) return;
    const int di = dst[i];
    const float v = __expf(out[i] - ord2f(segMaxU[di]));
    out[i] = v;
    atomicAdd(&segSum[di], v);
}

// ---------------- Kernel D: normalize ----------------
__global__ void __launch_bounds__(256)
gat_edge_norm(const int* __restrict__ dst, const float* __restrict__ segSum,
              float* __restrict__ out) {
    int i = blockIdx.x * blockDim.x + threadIdx.x;
    if (i >= NEDGES) return;
    out[i] = out[i] / segSum[dst[i]];
}

extern "C" void kernel_launch(void* const* d_in, const int* in_sizes, int n_in,
                              void* d_out, int out_size, void* d_ws, size_t ws_size,
                              hipStream_t stream) {
    const float* z_src = (const float*)d_in[0];
    const float* z_t   = (const float*)d_in[1];
    const float* z_c   = (const float*)d_in[2];
    const float* w     = (const float*)d_in[3];
    const int*   src   = (const int*)d_in[4];
    const int*   dst   = (const int*)d_in[5];
    float* out = (float*)d_out;

    // Workspace layout: p1 | p2 | segMaxU | segSum  (4 * NNODES * 4B = 800 KB)
    float*    p1      = (float*)d_ws;
    float*    p2      = p1 + NNODES;
    unsigned* segMaxU = (unsigned*)(p2 + NNODES);
    float*    segSum  = (float*)(segMaxU + NNODES);

    const dim3 blk(256);
    gat_node_proj<<<(NNODES + 255) / 256, blk, 0, stream>>>(
        z_src, z_t, w, p1, p2, segMaxU, segSum);

    // 16 edges per wave, 8 waves per block -> 128 edges per block, 6250 blocks exactly.
    gat_edge_score<<<NEDGES / 128, blk, 0, stream>>>(
        z_c, w + 2 * DIM, p1, p2, src, dst, out, segMaxU);

    gat_edge_exp<<<(NEDGES + 255) / 256, blk, 0, stream>>>(dst, segMaxU, out, segSum);

    gat_edge_norm<<<(NEDGES + 255) / 256, blk, 0, stream>>>(dst, segSum, out);
}